// ExpertChoiceRouter_57621281243492
// MI455X (gfx1250) — compile-verified
//
#include <hip/hip_runtime.h>
#include <hip/hip_bf16.h>

typedef __attribute__((ext_vector_type(2))) float v2f;
typedef __attribute__((ext_vector_type(8))) float v8f;

#define DIM   1024
#define NE    16
#define NB    16
#define SEQ   2048
#define NTOK  (NB * SEQ)      // 32768
#define CAP   2560            // int(1.25 * 32768 / 16)

// ---------------------------------------------------------------------------
// Kernel 1: logits[n][e] = sum_k x[n][k] * W[e][k]  via V_WMMA_F32_16X16X4_F32
// One wave handles a 16-token tile; 8 waves per block. W transposed into LDS
// (sWT[k*16+e]) for conflict-free B-operand reads.
// ---------------------------------------------------------------------------
__global__ void __launch_bounds__(256)
gemm_logits_wmma(const float* __restrict__ x, const float* __restrict__ W,
                 float* __restrict__ logits) {
  __shared__ float sWT[DIM * NE];  // 64 KB, transposed
  const int tid = threadIdx.x;
  for (int j = tid; j < DIM * NE; j += 256) {
    const int e = j >> 10, k = j & 1023;
    sWT[k * NE + e] = W[j];        // coalesced global read, strided LDS write
  }
  __syncthreads();

  const int lane  = tid & 31;
  const int wave  = tid >> 5;
  const int r16   = lane & 15;            // token row (A) / expert col (B)
  const int khalf = (lane >> 4) << 1;     // 0 for lanes 0-15, 2 for 16-31
  const long tileBase = ((long)blockIdx.x * 8 + wave) * 16;
  const float* xrow = x + (tileBase + r16) * DIM;

  v8f acc0 = {}, acc1 = {}, acc2 = {}, acc3 = {};
  for (int k = 0; k < DIM; k += 16) {
    {
      const int kk = k + khalf;
      v2f a = *(const v2f*)(xrow + kk);
      v2f b; b.x = sWT[kk * NE + r16]; b.y = sWT[(kk + 1) * NE + r16];
      acc0 = __builtin_amdgcn_wmma_f32_16x16x4_f32(false, a, false, b,
                                                   (short)0, acc0, false, false);
    }
    {
      const int kk = k + 4 + khalf;
      v2f a = *(const v2f*)(xrow + kk);
      v2f b; b.x = sWT[kk * NE + r16]; b.y = sWT[(kk + 1) * NE + r16];
      acc1 = __builtin_amdgcn_wmma_f32_16x16x4_f32(false, a, false, b,
                                                   (short)0, acc1, false, false);
    }
    {
      const int kk = k + 8 + khalf;
      v2f a = *(const v2f*)(xrow + kk);
      v2f b; b.x = sWT[kk * NE + r16]; b.y = sWT[(kk + 1) * NE + r16];
      acc2 = __builtin_amdgcn_wmma_f32_16x16x4_f32(false, a, false, b,
                                                   (short)0, acc2, false, false);
    }
    {
      const int kk = k + 12 + khalf;
      v2f a = *(const v2f*)(xrow + kk);
      v2f b; b.x = sWT[kk * NE + r16]; b.y = sWT[(kk + 1) * NE + r16];
      acc3 = __builtin_amdgcn_wmma_f32_16x16x4_f32(false, a, false, b,
                                                   (short)0, acc3, false, false);
    }
  }
  v8f c = (acc0 + acc1) + (acc2 + acc3);

  // C/D layout: VGPR v, lanes 0-15 -> M=v ; lanes 16-31 -> M=v+8 ; N = lane%16
  const int mBase = (lane >> 4) << 3;
  float* outp = logits + (tileBase + mBase) * NE + r16;
#pragma unroll
  for (int v = 0; v < 8; ++v) outp[v * NE] = c[v];
}

// ---------------------------------------------------------------------------
// Kernel 2: softmax over sequence axis, per (b, e) column of 2048 values.
// grid = (NE, NB); each thread holds 8 values in registers.
// ---------------------------------------------------------------------------
__global__ void __launch_bounds__(256)
softmax_seq(const float* __restrict__ logits, float* __restrict__ scores) {
  const int e = blockIdx.x;
  const int b = blockIdx.y;
  const int tid = threadIdx.x;
  const long base = (long)b * SEQ;

  float vals[8];
  float m = -3.402823466e38f;
#pragma unroll
  for (int j = 0; j < 8; ++j) {
    const int s = tid + j * 256;
    const float v = logits[(base + s) * NE + e];
    vals[j] = v;
    m = fmaxf(m, v);
  }

  __shared__ float red[256];
  red[tid] = m;
  __syncthreads();
  for (int off = 128; off > 0; off >>= 1) {
    if (tid < off) red[tid] = fmaxf(red[tid], red[tid + off]);
    __syncthreads();
  }
  const float gmax = red[0];
  __syncthreads();

  float sum = 0.f;
#pragma unroll
  for (int j = 0; j < 8; ++j) { vals[j] = __expf(vals[j] - gmax); sum += vals[j]; }
  red[tid] = sum;
  __syncthreads();
  for (int off = 128; off > 0; off >>= 1) {
    if (tid < off) red[tid] += red[tid + off];
    __syncthreads();
  }
  const float inv = 1.0f / red[0];
#pragma unroll
  for (int j = 0; j < 8; ++j) {
    const int s = tid + j * 256;
    scores[(base + s) * NE + e] = vals[j] * inv;
  }
}

// ---------------------------------------------------------------------------
// Kernel 3: exact k-th largest per expert via 4-pass byte radix-select on
// order-preserving float keys. One block per expert.
// ---------------------------------------------------------------------------
__device__ __forceinline__ unsigned fkey(float f) {
  const unsigned u = __float_as_uint(f);
  return (u & 0x80000000u) ? ~u : (u | 0x80000000u);
}

__global__ void __launch_bounds__(256)
radix_select_thr(const float* __restrict__ scores, unsigned* __restrict__ thr) {
  const int e = blockIdx.x;
  const int tid = threadIdx.x;
  __shared__ unsigned hist[256];
  __shared__ unsigned sBucket, sK;

  unsigned prefix = 0;
  int kth = CAP;  // 1-indexed rank from the top

  for (int pass = 3; pass >= 0; --pass) {
    const int shift = pass * 8;
    const unsigned mask = (pass == 3) ? 0u : (0xFFFFFFFFu << ((pass + 1) * 8));
    hist[tid] = 0;
    __syncthreads();

    for (int i = tid; i < NTOK; i += 256) {
      const unsigned u = fkey(scores[(long)i * NE + e]);
      if ((u & mask) == prefix) atomicAdd(&hist[(u >> shift) & 0xFF], 1u);
    }
    __syncthreads();

    if (tid == 0) {
      int rem = kth;
      int b = 255;
      for (; b > 0; --b) {
        const int c = (int)hist[b];
        if (rem <= c) break;
        rem -= c;
      }
      sBucket = (unsigned)b;
      sK = (unsigned)rem;
    }
    __syncthreads();
    prefix |= (sBucket << shift);
    kth = (int)sK;
    __syncthreads();  // before hist reset next pass
  }
  if (tid == 0) thr[e] = prefix;  // key of the CAP-th largest score of expert e
}

// ---------------------------------------------------------------------------
// Kernel 4: membership + overwrite scatter. For each token, highest expert e
// whose threshold it meets wins (matching the sequential scatter semantics).
// ---------------------------------------------------------------------------
__global__ void __launch_bounds__(256)
assign_out(const float* __restrict__ scores, const unsigned* __restrict__ thr,
           float* __restrict__ rw, int* __restrict__ ei) {
  __shared__ unsigned sThr[NE];
  if (threadIdx.x < NE) sThr[threadIdx.x] = thr[threadIdx.x];
  __syncthreads();

  const long i = (long)blockIdx.x * 256 + threadIdx.x;
  float w = 0.f;
  int idx = 0;
#pragma unroll
  for (int e = 0; e < NE; ++e) {
    const float s = scores[i * NE + e];  // contiguous: coalesced b128 loads
    if (fkey(s) >= sThr[e]) { w = s; idx = e; }
  }
  rw[i] = w;
  ei[i] = idx;
}

// ---------------------------------------------------------------------------
extern "C" void kernel_launch(void* const* d_in, const int* in_sizes, int n_in,
                              void* d_out, int out_size, void* d_ws, size_t ws_size,
                              hipStream_t stream) {
  const float* x = (const float*)d_in[0];  // [16, 2048, 1024] f32
  const float* W = (const float*)d_in[1];  // [16, 1024] f32

  float* rw     = (float*)d_out;                 // [N] routing weights
  int*   ei     = (int*)d_out + NTOK;            // [N] expert indices (i32 bits)
  float* scores = (float*)d_out + 2 * NTOK;      // [N, 16] softmax scores

  float*    logits = (float*)d_ws;                                     // 2 MB
  unsigned* thr    = (unsigned*)((char*)d_ws + (size_t)NTOK * NE * 4); // 64 B

  gemm_logits_wmma<<<NTOK / 128, 256, 0, stream>>>(x, W, logits);
  softmax_seq<<<dim3(NE, NB), 256, 0, stream>>>(logits, scores);
  radix_select_thr<<<NE, 256, 0, stream>>>(scores, thr);
  assign_out<<<NTOK / 256, 256, 0, stream>>>(scores, thr, rw, ei);
}